// FocusFormer_Attention_7980049236484
// MI455X (gfx1250) — compile-verified
//
#include <hip/hip_runtime.h>

// FocusFormer block for MI455X (gfx1250, wave32).
// 20 conv1x1 GEMMs (43 GFLOP) on the WMMA pipes via V_WMMA_F32_16X16X4_F32
// (exact fp32 to match the fp32 reference). Working set lives in the 192MB L2,
// so the limiter is matrix/issue throughput, not HBM (23.3 TB/s -> ~30us of
// traffic total). GEMM: 32x64 tile per wave (8 accumulators), with the shared
// B-tile cooperatively staged into LDS through the CDNA5 async copy path
// (global_load_async_to_lds_b128 + s_wait_asynccnt), double-buffered so the
// copy of K-chunk s+1 overlaps the WMMAs of K-chunk s.

#define Bsz  4
#define CDIM 256
#define Hdim 64
#define Wdim 64
#define HW   (Hdim * Wdim)              // 4096 spatial positions per image
#define CHW  (CDIM * HW)                // 1048576
#define TEN  ((size_t)Bsz * CHW)        // elements per [B,C,H,W] tensor

typedef __attribute__((ext_vector_type(2))) float v2f;
typedef __attribute__((ext_vector_type(8))) float v8f;

// ---------------------------------------------------------------------------
// conv1x1 as GEMM: Y[b,o,p] = sum_c W[o,c] * X[b,c,p]  (+bias[o]) (relu?)
// Block: 256 threads = 8 waves; block tile M=256 (wave w -> rows 32w..32w+31),
// N=64 columns. K walked in 8 chunks of 32 channels, each chunk async-copied
// to LDS once per block (8KB) and consumed by all 8 waves.
// WMMA f32 16x16x4 operand layouts (ISA 7.12.2):
//   A: lane%16 = M row; lane/16 selects K pair {0,1} vs {2,3}; v2f = 2 K vals.
//   B: lane%16 = N col; VGPR v + 2*(lane/16) = K row.
//   D: VGPR v + 8*(lane/16) = M row; lane%16 = N col.
// ---------------------------------------------------------------------------
__global__ __launch_bounds__(256) void conv1x1_wmma(
    const float* __restrict__ X,    // [B, C, HW]
    const float* __restrict__ Wm,   // [O=256, C=256] row-major
    const float* __restrict__ bias, // [256] or nullptr
    float* __restrict__ Y,          // [B, O, HW]
    int do_relu)
{
  __shared__ float sB[2][32 * 64];   // double-buffered K-chunk: 32 ch x 64 col

  const int tid  = threadIdx.x;
  const int lane = tid & 31;
  const int wave = tid >> 5;
  const int half = lane >> 4;        // 0: lanes 0-15, 1: lanes 16-31
  const int r    = lane & 15;

  const int n0 = blockIdx.x * 64;    // column base (spatial)
  const int b  = blockIdx.z;
  const int m0 = wave * 32;          // row base (out channel) for this wave

  const float* Xb = X + (size_t)b * CHW;
  float*       Yb = Y + (size_t)b * CHW;

  // Cooperative staging map: 512 16-byte units per 8KB chunk, 2 per thread.
  // unit u: channel = u>>4 (16 units of 4 cols per 64-col row), col = (u&15)*4
  const int ch0 = tid >> 4;          // channels 0..15
  const int ch1 = ch0 + 16;          // channels 16..31
  const int cu  = (tid & 15) << 2;   // column within tile (x4 floats)

#define STAGE(s_) do {                                                        \
    int k0_ = (s_) * 32;                                                      \
    unsigned l0_ = (unsigned)(size_t)&sB[(s_) & 1][ch0 * 64 + cu];            \
    unsigned l1_ = (unsigned)(size_t)&sB[(s_) & 1][ch1 * 64 + cu];            \
    const float* g0_ = Xb + (size_t)(k0_ + ch0) * HW + n0 + cu;               \
    const float* g1_ = Xb + (size_t)(k0_ + ch1) * HW + n0 + cu;               \
    asm volatile("global_load_async_to_lds_b128 %0, %1, off"                  \
                 :: "v"(l0_), "v"(g0_) : "memory");                           \
    asm volatile("global_load_async_to_lds_b128 %0, %1, off"                  \
                 :: "v"(l1_), "v"(g1_) : "memory");                           \
  } while (0)

  v8f accA[4] = {}, accB[4] = {};

  const float* wrow0 = Wm + (size_t)(m0 + r) * CDIM;      // rows m0..m0+15
  const float* wrow1 = wrow0 + 16 * CDIM;                 // rows m0+16..m0+31

  STAGE(0);
  asm volatile("s_wait_asynccnt 0x0" ::: "memory");
  __syncthreads();

  for (int s = 0; s < 8; ++s) {
    if (s < 7) STAGE(s + 1);         // overlap next copy with this compute

    const float* buf = sB[s & 1];
#pragma unroll
    for (int kk = 0; kk < 8; ++kk) { // 8 K-steps of 4 within the 32-chunk
      const int k = s * 32 + kk * 4;
      v2f a0, a1;
      a0.x = wrow0[k + 2 * half];  a0.y = wrow0[k + 2 * half + 1];
      a1.x = wrow1[k + 2 * half];  a1.y = wrow1[k + 2 * half + 1];
      const float* sb = buf + (kk * 4 + 2 * half) * 64;
#pragma unroll
      for (int j = 0; j < 4; ++j) {
        v2f bj;
        bj.x = sb[j * 16 + r];
        bj.y = sb[64 + j * 16 + r];
        accA[j] = __builtin_amdgcn_wmma_f32_16x16x4_f32(false, a0, false, bj,
                                                        (short)0, accA[j], false, false);
        accB[j] = __builtin_amdgcn_wmma_f32_16x16x4_f32(false, a1, false, bj,
                                                        (short)0, accB[j], false, false);
      }
    }

    if (s < 7) {
      // Stage s+1 must be fully in LDS (all waves) before consumption; the
      // same barrier also makes it safe for stage s+2 to overwrite buf[s&1].
      asm volatile("s_wait_asynccnt 0x0" ::: "memory");
      __syncthreads();
    }
  }
#undef STAGE

#pragma unroll
  for (int v = 0; v < 8; ++v) {
    int row0 = m0 + v + 8 * half;
    int row1 = row0 + 16;
    float bv0 = bias ? bias[row0] : 0.0f;
    float bv1 = bias ? bias[row1] : 0.0f;
#pragma unroll
    for (int j = 0; j < 4; ++j) {
      float oA = accA[j][v] + bv0;
      float oB = accB[j][v] + bv1;
      if (do_relu) { oA = fmaxf(oA, 0.0f); oB = fmaxf(oB, 0.0f); }
      Yb[(size_t)row0 * HW + n0 + j * 16 + r] = oA;
      Yb[(size_t)row1 * HW + n0 + j * 16 + r] = oB;
    }
  }
}

// ---------------------------------------------------------------------------
// Local 3x3 windowed attention per head (d=32, 8 heads). Zero-padded unfold
// semantics: out-of-bounds neighbors contribute logit == 0 (dot with zeros)
// and still participate in the softmax; their V contribution is 0.
// One thread per (b, head, y, x); x is the fastest index -> coalesced.
// ---------------------------------------------------------------------------
__global__ __launch_bounds__(256) void local_attn(
    const float* __restrict__ Q, const float* __restrict__ K,
    const float* __restrict__ V, float* __restrict__ O)
{
  int t    = blockIdx.x * 256 + threadIdx.x;  // 0 .. 4*8*64*64-1
  int x    = t & 63;
  int y    = (t >> 6) & 63;
  int head = (t >> 12) & 7;
  int b    = t >> 15;

  size_t base = (size_t)b * CHW + (size_t)head * 32 * HW;
  int p = y * Wdim + x;

  float q[32];
#pragma unroll
  for (int d = 0; d < 32; ++d) q[d] = Q[base + (size_t)d * HW + p];

  const float scale = 0.17677669529663687f;  // 1/sqrt(32)
  float logits[9];
#pragma unroll
  for (int nb = 0; nb < 9; ++nb) {
    int dy = nb / 3 - 1, dx = nb % 3 - 1;
    int yy = y + dy, xx = x + dx;
    float acc = 0.0f;
    if ((unsigned)yy < (unsigned)Hdim && (unsigned)xx < (unsigned)Wdim) {
      int pp = yy * Wdim + xx;
#pragma unroll
      for (int d = 0; d < 32; ++d) acc += q[d] * K[base + (size_t)d * HW + pp];
    }
    logits[nb] = acc * scale;
  }

  float mx = logits[0];
#pragma unroll
  for (int nb = 1; nb < 9; ++nb) mx = fmaxf(mx, logits[nb]);
  float s = 0.0f;
#pragma unroll
  for (int nb = 0; nb < 9; ++nb) { logits[nb] = expf(logits[nb] - mx); s += logits[nb]; }
  float inv = 1.0f / s;

  float out[32];
#pragma unroll
  for (int d = 0; d < 32; ++d) out[d] = 0.0f;
#pragma unroll
  for (int nb = 0; nb < 9; ++nb) {
    int dy = nb / 3 - 1, dx = nb % 3 - 1;
    int yy = y + dy, xx = x + dx;
    if ((unsigned)yy < (unsigned)Hdim && (unsigned)xx < (unsigned)Wdim) {
      int pp = yy * Wdim + xx;
      float a = logits[nb] * inv;
#pragma unroll
      for (int d = 0; d < 32; ++d) out[d] += a * V[base + (size_t)d * HW + pp];
    }
  }
#pragma unroll
  for (int d = 0; d < 32; ++d) O[base + (size_t)d * HW + p] = out[d];
}

// ---------------------------------------------------------------------------
// out[b,:,p] = LN_c( T[b,:,p] + R[b,:,p] ) * g + beta     (eps = 1e-6)
// One thread per spatial position; channel loop is coalesced across lanes.
// ---------------------------------------------------------------------------
__global__ __launch_bounds__(256) void layernorm_residual(
    const float* __restrict__ T, const float* __restrict__ R,
    const float* __restrict__ g, const float* __restrict__ beta,
    float* __restrict__ out)
{
  int q = blockIdx.x * 256 + threadIdx.x;  // 0 .. B*HW-1
  int b = q >> 12;
  int p = q & (HW - 1);
  size_t base = (size_t)b * CHW + p;

  float sum = 0.0f, sumsq = 0.0f;
  for (int c = 0; c < CDIM; ++c) {
    float v = T[base + (size_t)c * HW] + R[base + (size_t)c * HW];
    sum += v; sumsq += v * v;
  }
  float mu  = sum * (1.0f / CDIM);
  float var = fmaxf(sumsq * (1.0f / CDIM) - mu * mu, 0.0f);
  float inv = rsqrtf(var + 1e-6f);
  for (int c = 0; c < CDIM; ++c) {
    float v = T[base + (size_t)c * HW] + R[base + (size_t)c * HW];
    out[base + (size_t)c * HW] = (v - mu) * inv * g[c] + beta[c];
  }
}

// ---------------------------------------------------------------------------
// BatchNorm stats over (B,H,W) per channel. One block per channel.
// ---------------------------------------------------------------------------
__global__ __launch_bounds__(256) void bn_stats(
    const float* __restrict__ Y, float* __restrict__ mean, float* __restrict__ var)
{
  int c = blockIdx.x;
  float sum = 0.0f, sq = 0.0f;
  for (int b = 0; b < Bsz; ++b) {
    const float* p = Y + (size_t)b * CHW + (size_t)c * HW;
    for (int i = threadIdx.x; i < HW; i += 256) { float v = p[i]; sum += v; sq += v * v; }
  }
  __shared__ float ssum[256], ssq[256];
  ssum[threadIdx.x] = sum; ssq[threadIdx.x] = sq;
  __syncthreads();
  for (int s = 128; s > 0; s >>= 1) {
    if (threadIdx.x < s) {
      ssum[threadIdx.x] += ssum[threadIdx.x + s];
      ssq[threadIdx.x]  += ssq[threadIdx.x + s];
    }
    __syncthreads();
  }
  if (threadIdx.x == 0) {
    const float n = (float)(Bsz * HW);
    float m = ssum[0] / n;
    mean[c] = m;
    var[c]  = fmaxf(ssq[0] / n - m * m, 0.0f);
  }
}

// out = (Y - mu)/sqrt(var+1e-5) * g + beta + residual
__global__ __launch_bounds__(256) void bn_apply(
    const float* __restrict__ Y, const float* __restrict__ R,
    const float* __restrict__ mean, const float* __restrict__ var,
    const float* __restrict__ g, const float* __restrict__ beta,
    float* __restrict__ out)
{
  size_t i = (size_t)blockIdx.x * 256 + threadIdx.x;  // < TEN
  int c = (int)((i >> 12) & (CDIM - 1));
  out[i] = (Y[i] - mean[c]) * rsqrtf(var[c] + 1e-5f) * g[c] + beta[c] + R[i];
}

// ---------------------------------------------------------------------------
extern "C" void kernel_launch(void* const* d_in, const int* in_sizes, int n_in,
                              void* d_out, int out_size, void* d_ws, size_t ws_size,
                              hipStream_t stream) {
  const float* in1    = (const float*)d_in[0];
  const float* in2    = (const float*)d_in[1];
  const float* slf_wq = (const float*)d_in[2];
  const float* slf_wk = (const float*)d_in[3];
  const float* slf_wv = (const float*)d_in[4];
  const float* slf_fc = (const float*)d_in[5];
  const float* slf_g  = (const float*)d_in[6];
  const float* slf_b  = (const float*)d_in[7];
  const float* crs_wq = (const float*)d_in[8];
  const float* crs_wk = (const float*)d_in[9];
  const float* crs_wv = (const float*)d_in[10];
  const float* crs_fc = (const float*)d_in[11];
  const float* crs_g  = (const float*)d_in[12];
  const float* crs_b  = (const float*)d_in[13];
  const float* ffn_w1 = (const float*)d_in[14];
  const float* ffn_b1 = (const float*)d_in[15];
  const float* ffn_w2 = (const float*)d_in[16];
  const float* ffn_b2 = (const float*)d_in[17];
  const float* bn_g   = (const float*)d_in[18];
  const float* bn_b   = (const float*)d_in[19];
  // d_in[20]=kernel(3), d_in[21]=pad(1): compile-time constants here.

  float* ws = (float*)d_ws;
  float* Qb   = ws + 0 * TEN;
  float* Kb   = ws + 1 * TEN;
  float* Vb   = ws + 2 * TEN;   // reused as fc-GEMM output after attention
  float* Ob   = ws + 3 * TEN;
  float* SLF1 = ws + 4 * TEN;   // reused as CRS2 after SLF1 is consumed
  float* SLF2 = ws + 5 * TEN;
  float* CRS1 = ws + 6 * TEN;
  float* CRS2 = SLF1;
  float* mean = ws + 7 * TEN;
  float* var  = mean + CDIM;

  float* out1 = (float*)d_out;
  float* out2 = out1 + TEN;

  dim3 ggrid(HW / 64, 1, Bsz);   // (64, 1, 4): block tile = 256(M) x 64(N)

  auto gemm = [&](const float* X, const float* Wm, const float* bias, float* Y, int relu) {
    conv1x1_wmma<<<ggrid, 256, 0, stream>>>(X, Wm, bias, Y, relu);
  };

  auto mha = [&](const float* xq, const float* xkv,
                 const float* wq, const float* wk, const float* wv, const float* wfc,
                 const float* g, const float* beta, float* out) {
    gemm(xq,  wq, nullptr, Qb, 0);
    gemm(xkv, wk, nullptr, Kb, 0);
    gemm(xkv, wv, nullptr, Vb, 0);
    local_attn<<<(Bsz * 8 * HW) / 256, 256, 0, stream>>>(Qb, Kb, Vb, Ob);
    gemm(Ob, wfc, nullptr, Vb, 0);                       // Vb dead -> fc output
    layernorm_residual<<<(Bsz * HW) / 256, 256, 0, stream>>>(Vb, xq, g, beta, out);
  };

  auto ffn = [&](const float* x, float* out) {
    gemm(x,  ffn_w1, ffn_b1, Qb, 1);                     // relu fused
    gemm(Qb, ffn_w2, ffn_b2, Kb, 0);
    bn_stats<<<CDIM, 256, 0, stream>>>(Kb, mean, var);
    bn_apply<<<(unsigned)(TEN / 256), 256, 0, stream>>>(Kb, x, mean, var, bn_g, bn_b, out);
  };

  mha(in1,  in1,  slf_wq, slf_wk, slf_wv, slf_fc, slf_g, slf_b, SLF1);
  mha(in2,  in2,  slf_wq, slf_wk, slf_wv, slf_fc, slf_g, slf_b, SLF2);
  mha(SLF1, SLF2, crs_wq, crs_wk, crs_wv, crs_fc, crs_g, crs_b, CRS1);
  mha(SLF2, SLF1, crs_wq, crs_wk, crs_wv, crs_fc, crs_g, crs_b, CRS2);
  ffn(CRS1, out1);
  ffn(CRS2, out2);
}